// CTCLayer_55370718380492
// MI455X (gfx1250) — compile-verified
//
#include <hip/hip_runtime.h>
#include <stdint.h>

#define B_   512
#define T_   512
#define C_   128
#define L_   48
#define S_   97          // 2*L+1
#define NBUF 48          // async ring depth (<=63: ASYNCcnt is 6 bits)
#define EPS_ 1e-7f

// Issue one async 512B row copy (32 lanes x B128) global -> LDS ring slot.
// NT hint: y_pred is streamed once, never reused -> bypass temporal caching.
__device__ __forceinline__ void async_row(uint32_t lds, uint32_t voff,
                                          const float* base) {
    asm volatile("global_load_async_to_lds_b128 %0, %1, %2 th:TH_LOAD_NT"
                 :: "v"(lds), "v"(voff), "s"(base) : "memory");
}

// One wave32 per batch element. Lane l owns lattice states s = 4l..4l+3.
__global__ __launch_bounds__(32) void ctc_fwd_kernel(const int* __restrict__ yt,
                                                     const float* __restrict__ yp,
                                                     float* __restrict__ out) {
    __shared__ float buf[NBUF * C_];          // 48 time-rows of 128 probs = 24 KB

    const int b    = blockIdx.x;
    const int lane = threadIdx.x;

    const float* rowbase = yp + (size_t)b * T_ * C_;          // uniform -> SGPR pair
    const uint32_t ldsBase = (uint32_t)(uintptr_t)&buf[0];    // LDS byte offset
    const uint32_t laneB   = (uint32_t)(lane * 16);           // 16B per lane

    // ---- prologue: queue NBUF async row loads ----
    for (int r = 0; r < NBUF; ++r) {
        const uint32_t off = (uint32_t)(r * C_ * 4) + laneB;
        async_row(ldsBase + off, off, rowbase);
    }

    // ---- per-lane lattice metadata ----
    // k even -> blank (class 127). k=1 -> label idx 2l, k=3 -> label idx 2l+1.
    int   e1 = C_ - 1, e3 = C_ - 1;
    float skip1 = 0.f, skip3 = 0.f;
    const bool act1 = (4 * lane + 1) < S_;    // lanes 0..23
    const bool act3 = (4 * lane + 3) < S_;    // lanes 0..23
    int lab1 = 0, lab3 = 0, lab0 = 0;
    if (act1)             lab1 = yt[b * L_ + 2 * lane];
    if (act3)             lab3 = yt[b * L_ + 2 * lane + 1];
    if (act1 && lane > 0) lab0 = yt[b * L_ + 2 * lane - 1];
    if (act1) { e1 = lab1; if (lane > 0) skip1 = (lab1 != lab0) ? 1.f : 0.f; }
    if (act3) { e3 = lab3; skip3 = (lab3 != lab1) ? 1.f : 0.f; }
    const float act0f = (4 * lane     < S_) ? 1.f : 0.f;      // lanes 0..24
    const float act1f = act1 ? 1.f : 0.f;
    const float act2f = (4 * lane + 2 < S_) ? 1.f : 0.f;
    const float act3f = act3 ? 1.f : 0.f;

    float a0, a1, a2, a3;                     // alpha (linear domain)
    float logc = 0.f;                         // accumulated log-normalizer

    // ---- t = 0 (peeled): alpha0[s] = p0[s] for s<2 else 0 ----
    {
        asm volatile("s_wait_asynccnt %0" :: "n"(NBUF - 1) : "memory");
        const float pb = buf[C_ - 1] + EPS_;
        const float p1 = buf[e1]     + EPS_;
        a0 = (lane == 0) ? pb : 0.f;
        a1 = (lane == 0) ? p1 : 0.f;
        a2 = 0.f; a3 = 0.f;
        // refill slot 0 with row NBUF (reads above drained via s_wait_dscnt)
        const uint32_t off = laneB;                           // slot 0
        asm volatile("s_wait_dscnt 0" ::: "memory");
        async_row(ldsBase + off, (uint32_t)(NBUF * C_ * 4) + laneB, rowbase);
    }

    // ---- main recurrence t = 1..T-1 ----
    for (int t = 1; t < T_; ++t) {
        // row t complete once <= NBUF-1 newer loads remain outstanding
        asm volatile("s_wait_asynccnt %0" :: "n"(NBUF - 1) : "memory");

        const int slot = (t % NBUF) * C_;
        const float pb = buf[slot + (C_ - 1)] + EPS_;   // blank (broadcast)
        const float p1 = buf[slot + e1]       + EPS_;
        const float p3 = buf[slot + e3]       + EPS_;

        // refill this ring slot (clamped at T-1: redundant identical rewrites
        // keep the wait invariant constant through the tail). s_wait_dscnt 0
        // guarantees the ds reads above drained before the slot is overwritten.
        {
            int r = t + NBUF; if (r > T_ - 1) r = T_ - 1;
            const uint32_t lds  = ldsBase + (uint32_t)((r % NBUF) * C_ * 4) + laneB;
            const uint32_t voff = (uint32_t)(r * C_ * 4) + laneB;
            asm volatile("s_wait_dscnt 0" ::: "memory");
            async_row(lds, voff, rowbase);
        }

        // single cross-lane term per step: prev lane's a3
        // (state s-1 for k=0, state s-2 for k=1)
        float pa3 = __shfl_up(a3, 1, 32);
        if (lane == 0) pa3 = 0.f;
        const float n0 = (a0 + pa3)              * pb * act0f;  // blank
        const float n1 = (a1 + a0 + skip1 * pa3) * p1 * act1f;  // label
        const float n2 = (a2 + a1)               * pb * act2f;  // blank
        const float n3 = (a3 + a2 + skip3 * a1)  * p3 * act3f;  // label
        a0 = n0; a1 = n1; a2 = n2; a3 = n3;

        // renormalize every 8 steps; fold scale into logc
        if ((t & 7) == 7) {
            float z = (a0 + a1) + (a2 + a3);
            for (int off = 16; off > 0; off >>= 1) z += __shfl_xor(z, off, 32);
            const float r = __builtin_amdgcn_rcpf(z);
            a0 *= r; a1 *= r; a2 *= r; a3 *= r;
            logc += __logf(z);
        }
    }

    // loss = -( log(A[96] + A[95]) + logc );  s=96 -> lane24.k0, s=95 -> lane23.k3
    const float v96 = __shfl(a0, 24, 32);
    const float v95 = __shfl(a3, 23, 32);
    if (lane == 0) out[b] = -(__logf(v95 + v96) + logc);
}

extern "C" void kernel_launch(void* const* d_in, const int* in_sizes, int n_in,
                              void* d_out, int out_size, void* d_ws, size_t ws_size,
                              hipStream_t stream) {
    const int*   yt  = (const int*)d_in[0];    // y_true [512,48] int32
    const float* yp  = (const float*)d_in[1];  // y_pred [512,512,128] f32
    float*       out = (float*)d_out;          // [512,1] f32
    hipLaunchKernelGGL(ctc_fwd_kernel, dim3(B_), dim3(32), 0, stream, yt, yp, out);
}